// Criterion_49237505081886
// MI455X (gfx1250) — compile-verified
//
#include <hip/hip_runtime.h>

typedef float v2f __attribute__((ext_vector_type(2)));
typedef float v8f __attribute__((ext_vector_type(8)));

#define LOG_2PI_F 1.8378770664093453f
#define NT 2048

// One workgroup (8 wave32s). Reads ONLY the last row of each (bs, nt) array:
// 24 KB total -> pure latency-bound; everything happens in a single WGP.
__global__ __launch_bounds__(256) void criterion_lastrow_kernel(
    const float* __restrict__ mu, const float* __restrict__ sigma,
    const float* __restrict__ ty, float* __restrict__ out, int bs) {
  __shared__ float partials[256];
  const int tid = threadIdx.x;
  const size_t row = (size_t)(bs - 1) * (size_t)NT;

  const float4* __restrict__ m4 = (const float4*)(mu + row);
  const float4* __restrict__ s4 = (const float4*)(sigma + row);
  const float4* __restrict__ y4 = (const float4*)(ty + row);

  // 2048 elements / 256 threads = 8 per thread = 2x float4 per array
  // (emits global_load_b128; keeps 6 loads in flight per thread).
  float acc = 0.0f;
#pragma unroll
  for (int j = 0; j < 2; ++j) {
    const int idx = tid * 2 + j;  // float4 index, 0..511
    float4 m = m4[idx];
    float4 s = s4[idx];
    float4 y = y4[idx];
    float dx = y.x - m.x, dy = y.y - m.y, dz = y.z - m.z, dw = y.w - m.w;
    acc += dx * dx / s.x + __logf(s.x);
    acc += dy * dy / s.y + __logf(s.y);
    acc += dz * dz / s.z + __logf(s.z);
    acc += dw * dw / s.w + __logf(s.w);
  }
  partials[tid] = acc;
  __syncthreads();

  // Wave 0 folds the 256 partials on the matrix pipe:
  // D = A x ones(4x16) + C accumulates row-sums of A; 4 WMMAs consume
  // 64 partials each (2 per lane into the 16x4 f32 A tile).
  if (tid < 32) {  // wave-uniform branch: EXEC is all-ones inside (WMMA req.)
    v8f c = {0.f, 0.f, 0.f, 0.f, 0.f, 0.f, 0.f, 0.f};
    v2f ones;
    ones[0] = 1.0f;
    ones[1] = 1.0f;
#pragma unroll
    for (int j = 0; j < 4; ++j) {
      v2f a;
      a[0] = partials[j * 64 + tid * 2 + 0];
      a[1] = partials[j * 64 + tid * 2 + 1];
      // (neg_a, A, neg_b, B, c_mod, C, reuse_a, reuse_b)
      c = __builtin_amdgcn_wmma_f32_16x16x4_f32(false, a, false, ones,
                                                (short)0, c, false, false);
    }
    // D layout: lanes 0-15 hold rows m=0..7 in VGPR 0..7, lanes 16-31 rows
    // m=8..15. Per-lane 8-way add, then fold the two lane halves.
    float s = c[0] + c[1] + c[2] + c[3] + c[4] + c[5] + c[6] + c[7];
    s += __shfl_xor(s, 16, 32);
    if (tid == 0) {
      float total = s + (float)NT * LOG_2PI_F;
      out[0] = 0.5f * total / ((float)NT * (float)bs);
    }
  }
}

extern "C" void kernel_launch(void* const* d_in, const int* in_sizes, int n_in,
                              void* d_out, int out_size, void* d_ws,
                              size_t ws_size, hipStream_t stream) {
  const float* mu = (const float*)d_in[0];
  const float* sigma = (const float*)d_in[1];
  const float* ty = (const float*)d_in[2];
  float* out = (float*)d_out;
  const int bs = in_sizes[0] / NT;  // reference: (4096, 2048)

  criterion_lastrow_kernel<<<1, 256, 0, stream>>>(mu, sigma, ty, out, bs);
}